// Fuzzy_MultiHeadAttention_DTW_69200513073605
// MI455X (gfx1250) — compile-verified
//
#include <hip/hip_runtime.h>
#include <hip/hip_bf16.h>

// ---------------------------------------------------------------------------
// Fuzzy MHA + softDTW on MI455X (gfx1250, wave32).
//   K1: vals = value @ W_values^T          -- V_WMMA_F32_16X16X4_F32
//   K2: soft-DTW anti-diagonal wavefront   -- 1 block / (bn,r), LDS diagonals
//   K3: softmax(R=16) + weighted mix       -- tiny
//   K4: out  = out0 @ W_fc^T + b_fc        -- V_WMMA_F32_16X16X4_F32
// ---------------------------------------------------------------------------

typedef __attribute__((ext_vector_type(2))) float v2f;
typedef __attribute__((ext_vector_type(8))) float v8f;

#define BIGF 1e30f

// ---------------------------------------------------------------------------
// K1: C[m][n] = sum_k value[m][k] * Wv[n][k]   (M=256, N=2048, K=128)
// One wave per 16x16 tile. A lane layout (f32 16x4): lane half selects K pair.
// ---------------------------------------------------------------------------
__global__ void gemm_vals_kernel(const float* __restrict__ value,
                                 const float* __restrict__ Wv,
                                 float* __restrict__ vals) {
    const int wave = (blockIdx.x * blockDim.x + threadIdx.x) >> 5;
    const int lane = threadIdx.x & 31;
    const int nt = wave & 127;          // 128 N-tiles
    const int mt = wave >> 7;           // 16  M-tiles
    const int m0 = mt * 16, n0 = nt * 16;
    const int half = lane >> 4;         // K-pair select
    const int row  = lane & 15;         // M (for A) / N (for B)

    const float* __restrict__ arow = value + (m0 + row) * 128;
    const float* __restrict__ brow = Wv    + (n0 + row) * 128; // B[k][n]=Wv[n][k]

    v8f c = {};
    for (int k0 = 0; k0 < 128; k0 += 4) {
        v2f a, b;
        a.x = arow[k0 + 2 * half];
        a.y = arow[k0 + 2 * half + 1];
        b.x = brow[k0 + 2 * half];
        b.y = brow[k0 + 2 * half + 1];
        c = __builtin_amdgcn_wmma_f32_16x16x4_f32(false, a, false, b,
                                                  (short)0, c, false, false);
    }
#pragma unroll
    for (int v = 0; v < 8; ++v)
        vals[(m0 + v + 8 * half) * 2048 + n0 + row] = c[v];
}

// ---------------------------------------------------------------------------
// K2: soft-DTW, gamma=1, squared-euclidean 1-d cost. Grid R[(T+1)x(T+1)],
// T=128. Block = 128 threads, thread tid owns DP row i=tid+1. Three rotating
// LDS anti-diagonal buffers; one barrier per diagonal (255 diagonals).
// ---------------------------------------------------------------------------
__global__ void softdtw_kernel(const float* __restrict__ q,
                               const float* __restrict__ keys,
                               float* __restrict__ dist) {
    const int p  = blockIdx.x;          // 0..4095
    const int bn = p >> 4;
    const int r  = p & 15;
    const int tid = threadIdx.x;        // 0..127
    const int i   = tid + 1;            // DP row

    __shared__ float yrow[128];
    __shared__ float buf[3][129];

    const float xv = q[bn * 128 + tid]; // x[i-1]
    yrow[tid] = keys[r * 128 + tid];

    // diag d=0: only (0,0)=0 exists; diag d=1: all borders = BIG
    buf[0][i] = BIGF;
    buf[1][i] = BIGF;
    if (tid == 0) { buf[0][0] = 0.0f; buf[1][0] = BIGF; }
    __syncthreads();

    int ia = 0, ib = 1, ic = 2;         // d-2, d-1, current
    float cur = BIGF;
    for (int d = 2; d <= 256; ++d) {
        const int j = d - i;
        const float p2  = buf[ia][i - 1];   // R[i-1][j-1]
        const float p1a = buf[ib][i - 1];   // R[i-1][j]
        const float p1b = buf[ib][i];       // R[i][j-1]
        if (j >= 1 && j <= 128) {
            const float diff = xv - yrow[j - 1];
            float m = fminf(fminf(p2, p1a), p1b);
            float s = __expf(m - p2) + __expf(m - p1a) + __expf(m - p1b);
            cur = diff * diff + (m - __logf(s));
        } else {
            cur = BIGF;
        }
        buf[ic][i] = cur;
        if (tid == 0) buf[ic][0] = BIGF;    // row-0 border of new diagonal
        __syncthreads();
        const int t = ia; ia = ib; ib = ic; ic = t;
    }
    // d=256 is valid only for i=128 (tid 127): R[T][T]
    if (tid == 127) dist[p] = cur;
}

// ---------------------------------------------------------------------------
// K3: z = -0.5*(dist/width)^2 ; softmax over R=16 ; out0 = attn . vals
// One block per bn (256 blocks), one thread per feature d (128 threads).
// ---------------------------------------------------------------------------
__global__ void attn_mix_kernel(const float* __restrict__ dist,
                                const float* __restrict__ widths,
                                const float* __restrict__ vals,
                                float* __restrict__ out0) {
    const int bn  = blockIdx.x;
    const int tid = threadIdx.x;        // 0..127
    __shared__ float attnS[16];

    if (tid < 16) {
        const float dv = dist[bn * 16 + tid] / widths[tid];
        attnS[tid] = -0.5f * dv * dv;
    }
    __syncthreads();
    if (tid == 0) {
        float mx = attnS[0];
        for (int r = 1; r < 16; ++r) mx = fmaxf(mx, attnS[r]);
        float e[16], s = 0.0f;
        for (int r = 0; r < 16; ++r) { e[r] = __expf(attnS[r] - mx); s += e[r]; }
        const float inv = 1.0f / s;
        for (int r = 0; r < 16; ++r) attnS[r] = e[r] * inv;
    }
    __syncthreads();

    float acc = 0.0f;
    const float* __restrict__ vrow = vals + bn * 2048 + tid;
#pragma unroll
    for (int r = 0; r < 16; ++r) acc += attnS[r] * vrow[r * 128];
    out0[bn * 128 + tid] = acc;
}

// ---------------------------------------------------------------------------
// K4: out[m][n] = sum_k out0[m][k] * Wfc[n][k] + bfc[n]   (M=256,N=128,K=128)
// ---------------------------------------------------------------------------
__global__ void gemm_fc_kernel(const float* __restrict__ out0,
                               const float* __restrict__ Wfc,
                               const float* __restrict__ bfc,
                               float* __restrict__ out) {
    const int wave = (blockIdx.x * blockDim.x + threadIdx.x) >> 5;
    const int lane = threadIdx.x & 31;
    const int nt = wave & 7;            // 8 N-tiles
    const int mt = wave >> 3;           // 16 M-tiles
    const int m0 = mt * 16, n0 = nt * 16;
    const int half = lane >> 4;
    const int row  = lane & 15;

    const float* __restrict__ arow = out0 + (m0 + row) * 128;
    const float* __restrict__ brow = Wfc  + (n0 + row) * 128;

    v8f c = {};
    for (int k0 = 0; k0 < 128; k0 += 4) {
        v2f a, b;
        a.x = arow[k0 + 2 * half];
        a.y = arow[k0 + 2 * half + 1];
        b.x = brow[k0 + 2 * half];
        b.y = brow[k0 + 2 * half + 1];
        c = __builtin_amdgcn_wmma_f32_16x16x4_f32(false, a, false, b,
                                                  (short)0, c, false, false);
    }
    const float bias = bfc[n0 + row];
#pragma unroll
    for (int v = 0; v < 8; ++v)
        out[(m0 + v + 8 * half) * 128 + n0 + row] = c[v] + bias;
}

// ---------------------------------------------------------------------------
// launch
// inputs: 0=query 1=key(unused) 2=value 3=W_values 4=keys_param
//         5=rules_widths 6=W_fc 7=b_fc
// ---------------------------------------------------------------------------
extern "C" void kernel_launch(void* const* d_in, const int* in_sizes, int n_in,
                              void* d_out, int out_size, void* d_ws, size_t ws_size,
                              hipStream_t stream) {
    const float* query  = (const float*)d_in[0];
    const float* value  = (const float*)d_in[2];
    const float* Wv     = (const float*)d_in[3];
    const float* keysp  = (const float*)d_in[4];
    const float* widths = (const float*)d_in[5];
    const float* Wfc    = (const float*)d_in[6];
    const float* bfc    = (const float*)d_in[7];
    float* out = (float*)d_out;

    char* ws = (char*)d_ws;
    float* vals = (float*)ws;                                   // 256*2048 f32 (2 MB)
    float* dist = (float*)(ws + (size_t)256 * 2048 * 4);        // 4096 f32
    float* out0 = (float*)(ws + (size_t)256 * 2048 * 4 + 4096 * 4); // 256*128 f32

    // K1: 2048 tiles, 1 wave each, 8 waves/block -> 256 blocks x 256 thr
    gemm_vals_kernel<<<256, 256, 0, stream>>>(value, Wv, vals);
    // K2: 4096 DP problems, 128 threads each
    softdtw_kernel<<<4096, 128, 0, stream>>>(query, keysp, dist);
    // K3: 256 rows
    attn_mix_kernel<<<256, 128, 0, stream>>>(dist, widths, vals, out0);
    // K4: 128 tiles -> 16 blocks x 256 thr
    gemm_fc_kernel<<<16, 256, 0, stream>>>(out0, Wfc, bfc, out);
}